// Model_Fidelity_9844065042870
// MI455X (gfx1250) — compile-verified
//
#include <hip/hip_runtime.h>
#include <hip/hip_bf16.h>
#include <math.h>
#include <float.h>

// ---------------------------------------------------------------------------
// Fused fidelity kernel for MI455X (gfx1250, wave32).
//   fidelity = |<psi2|psi1>|^2,  psi_i = embed(mlp(x_i)) |0000>
// MLP on matrix cores via V_WMMA_F32_16X16X4_F32 (D = W^T(A) x X^T(B) + bias(C)).
// Circuit: per layer  U = [H on all 4 wires] then [ONE fused diagonal]
// (RZ's commute past later H's; CNOT.RZ.CNOT is diagonal; angles are
// layer-invariant so the 16 unit phases are precomputed once per embed).
// All weight/bias masking is arithmetic (v_cndmask/v_mul), never conditional
// loads, so EXEC stays full and no exec-save branches are generated.
// ReLU uses v_med3_f32 (single instruction, self-canonicalizing).
// ---------------------------------------------------------------------------

typedef float v2f __attribute__((ext_vector_type(2)));
typedef float v8f __attribute__((ext_vector_type(8)));

#define INV_SQRT2_F 0.70710678118654752440f
#define PI_F        3.14159265358979323846f

__device__ __forceinline__ float relu1(float x) {
  // v_med3_f32 x, 0, +FLT_MAX  ->  ReLU in one VALU op (quiets NaN itself)
  return __builtin_amdgcn_fmed3f(x, 0.0f, FLT_MAX);
}

__device__ __forceinline__ v8f wmma16x16x4(v2f a, v2f b, v8f c) {
  // (neg_a, A, neg_b, B, c_mod, C, reuse_a, reuse_b)
  return __builtin_amdgcn_wmma_f32_16x16x4_f32(false, a, false, b, (short)0, c,
                                               false, false);
}

// Per-wave constants: all WMMA A-operands (transposed, zero-padded weights)
// and C-operands (bias broadcast) -- identical for all four MLP chains.
// f32 16x4 A layout: (m,k) -> vgpr (k&1), lane m + 16*(k>>1)
// f32 4x16 B layout: (k,n) -> vgpr (k&1), lane n + 16*(k>>1)
// f32 16x16 C layout: (m,n) -> vgpr m&7, lane n + 16*(m>>3)
struct MlpConsts {
  v2f a1;          // W1^T  (16x4 pad), K slice 0..3
  v2f a2_0, a2_1;  // W2^T  K slices 0..3 / 4..7
  v2f a3_0, a3_1;  // W3^T  K slices 0..3 / 4..7
  v8f c1, c2, c3;  // bias C operands
};

__device__ __forceinline__ void build_consts(
    const float* __restrict__ W1, const float* __restrict__ b1,
    const float* __restrict__ W2, const float* __restrict__ b2,
    const float* __restrict__ W3, const float* __restrict__ b3,
    int lane, MlpConsts& mc)
{
  const int m  = lane & 15;
  const int kh = lane >> 4;
  const int k0 = 2 * kh;
  const int m8 = m & 7, m4 = m & 3;
  // arithmetic masks -> v_cndmask once, then v_mul; loads stay unconditional
  const float mask8 = (m < 8)   ? 1.0f : 0.0f;
  const float mask4 = (m < 4)   ? 1.0f : 0.0f;
  const float maskK = (kh == 0) ? 1.0f : 0.0f;

  mc.a1.x   = W1[(k0 + 0) * 8 + m8] * mask8;   // W1 is (4,8) row-major
  mc.a1.y   = W1[(k0 + 1) * 8 + m8] * mask8;
  mc.a2_0.x = W2[(0 + k0 + 0) * 8 + m8] * mask8;  // W2 is (8,8)
  mc.a2_0.y = W2[(0 + k0 + 1) * 8 + m8] * mask8;
  mc.a2_1.x = W2[(4 + k0 + 0) * 8 + m8] * mask8;
  mc.a2_1.y = W2[(4 + k0 + 1) * 8 + m8] * mask8;
  mc.a3_0.x = W3[(0 + k0 + 0) * 4 + m4] * mask4;  // W3 is (8,4)
  mc.a3_0.y = W3[(0 + k0 + 1) * 4 + m4] * mask4;
  mc.a3_1.x = W3[(4 + k0 + 0) * 4 + m4] * mask4;
  mc.a3_1.y = W3[(4 + k0 + 1) * 4 + m4] * mask4;
#pragma unroll
  for (int v = 0; v < 8; ++v) {
    mc.c1[v] = b1[v] * maskK;                  // uniform s_load + v_mul
    mc.c2[v] = b2[v] * maskK;
    mc.c3[v] = (v < 4) ? (b3[v] * maskK) : 0.0f;   // v is compile-time
  }
}

// One 16-row MLP tile: 5 WMMAs. Activations stay in B/C roles; the C->B
// layout conversion per K=4 slice is two ds_bpermutes (__shfl) + cndmasks.
__device__ __forceinline__ void mlp_tile(const float* __restrict__ x, int row0,
                                         const MlpConsts& mc, int lane,
                                         float fout[4])
{
  const int m  = lane & 15;
  const int kh = lane >> 4;
  const int k0 = 2 * kh;

  // B operand of layer 1 = X^T: one 8B load/lane, every x byte read once
  const v2f bx = *(const v2f*)(x + (size_t)(row0 + m) * 4 + k0);

  v8f h1 = wmma16x16x4(mc.a1, bx, mc.c1);
#pragma unroll
  for (int v = 0; v < 8; ++v) h1[v] = relu1(h1[v]);

  v8f acc = mc.c2;
#pragma unroll
  for (int s = 0; s < 2; ++s) {
    const int kb = 4 * s;
    v2f b;
    const float t0 = __shfl(h1[kb + 2], m, 32);
    const float t1 = __shfl(h1[kb + 3], m, 32);
    b.x = (kh == 0) ? h1[kb + 0] : t0;
    b.y = (kh == 0) ? h1[kb + 1] : t1;
    acc = wmma16x16x4((s == 0) ? mc.a2_0 : mc.a2_1, b, acc);
  }
  v8f h2 = acc;
#pragma unroll
  for (int v = 0; v < 8; ++v) h2[v] = relu1(h2[v]);

  acc = mc.c3;
#pragma unroll
  for (int s = 0; s < 2; ++s) {
    const int kb = 4 * s;
    v2f b;
    const float t0 = __shfl(h2[kb + 2], m, 32);
    const float t1 = __shfl(h2[kb + 3], m, 32);
    b.x = (kh == 0) ? h2[kb + 0] : t0;
    b.y = (kh == 0) ? h2[kb + 1] : t1;
    acc = wmma16x16x4((s == 0) ? mc.a3_0 : mc.a3_1, b, acc);
  }
#pragma unroll
  for (int v = 0; v < 4; ++v) fout[v] = acc[v];  // lanes 0..15: f[row0+n][v]
}

// Forward embed of |0000>. Wire w bit of index i is (i & (8>>w)).
// Combined per-layer diagonal: theta_i = sum_j s_j(i)*(-f_j)
//                                      + sum_p s_p(i)*(-(pi-f_a)(pi-f_b)),
// s_j = +-1 from bit_j, s_p = +-1 from bit_a XOR bit_b. Layer-invariant ->
// 16 sincos per embed, reused by all 3 layers. Layer-1 H-block on |0000> is
// the uniform +1/4 state, so only 2 butterfly blocks remain.
__device__ __forceinline__ void embed_state(const float f[4],
                                            float sr[16], float si[16])
{
  float alpha[4], beta[4];
#pragma unroll
  for (int j = 0; j < 4; ++j) alpha[j] = -f[j];
#pragma unroll
  for (int p = 0; p < 4; ++p)
    beta[p] = -(PI_F - f[p]) * (PI_F - f[(p + 1) & 3]);

  float cp[16], sp[16];
#pragma unroll
  for (int i = 0; i < 16; ++i) {
    float th = 0.0f;
#pragma unroll
    for (int j = 0; j < 4; ++j)
      th += (i & (8 >> j)) ? alpha[j] : -alpha[j];
#pragma unroll
    for (int p = 0; p < 4; ++p) {
      const bool d = ((i & (8 >> p)) != 0) != ((i & (8 >> ((p + 1) & 3))) != 0);
      th += d ? beta[p] : -beta[p];
    }
    __sincosf(th, &sp[i], &cp[i]);
  }

  // layer 1: H-block(|0000>) = uniform 1/4, then fused diagonal
#pragma unroll
  for (int i = 0; i < 16; ++i) {
    sr[i] = 0.25f * cp[i];
    si[i] = 0.25f * sp[i];
  }

  // layers 2..3 (rolled: 2x smaller static code, same dynamic work;
  // all body indices are constants so state stays in VGPRs)
#pragma unroll 1
  for (int layer = 1; layer < 3; ++layer) {
#pragma unroll
    for (int j = 0; j < 4; ++j) {
      const int st = 8 >> j;
#pragma unroll
      for (int i = 0; i < 16; ++i) {
        if (i & st) continue;
        const int i2 = i | st;
        const float ar = sr[i],  ai = si[i];
        const float br = sr[i2], bi = si[i2];
        sr[i]  = (ar + br) * INV_SQRT2_F;
        si[i]  = (ai + bi) * INV_SQRT2_F;
        sr[i2] = (ar - br) * INV_SQRT2_F;
        si[i2] = (ai - bi) * INV_SQRT2_F;
      }
    }
#pragma unroll
    for (int i = 0; i < 16; ++i) {
      const float re = sr[i], im = si[i];
      sr[i] = re * cp[i] - im * sp[i];
      si[i] = im * cp[i] + re * sp[i];
    }
  }
}

__global__ __launch_bounds__(256)
void fidelity_kernel(const float* __restrict__ x1, const float* __restrict__ x2,
                     const float* __restrict__ W1, const float* __restrict__ b1,
                     const float* __restrict__ W2, const float* __restrict__ b2,
                     const float* __restrict__ W3, const float* __restrict__ b3,
                     float* __restrict__ out, int Btot)
{
  const int lane = threadIdx.x & 31;
  const int wave = blockIdx.x * (blockDim.x >> 5) + (threadIdx.x >> 5);
  const int base = wave * 32;              // 32 rows per wave
  if (base + 32 > Btot) return;            // wave-uniform guard (EXEC stays full)

  MlpConsts mc;
  build_consts(W1, b1, W2, b2, W3, b3, lane, mc);

  float fa[4], fb[4], f1[4], f2[4];

  // x1: two 16-row WMMA tiles, then spread tile1 results into lanes 16..31
  mlp_tile(x1, base,      mc, lane, fa);
  mlp_tile(x1, base + 16, mc, lane, fb);
#pragma unroll
  for (int v = 0; v < 4; ++v) {
    const float t = __shfl(fb[v], lane & 15, 32);
    f1[v] = (lane < 16) ? fa[v] : t;
  }
  // x2
  mlp_tile(x2, base,      mc, lane, fa);
  mlp_tile(x2, base + 16, mc, lane, fb);
#pragma unroll
  for (int v = 0; v < 4; ++v) {
    const float t = __shfl(fb[v], lane & 15, 32);
    f2[v] = (lane < 16) ? fa[v] : t;
  }

  // Register-resident 4-qubit simulation (one row per lane)
  float r1[16], i1[16], r2[16], i2[16];
  embed_state(f1, r1, i1);
  embed_state(f2, r2, i2);

  // fidelity = |<psi2|psi1>|^2
  float dre = 0.0f, dim = 0.0f;
#pragma unroll
  for (int i = 0; i < 16; ++i) {
    dre += r2[i] * r1[i] + i2[i] * i1[i];
    dim += r2[i] * i1[i] - i2[i] * r1[i];
  }
  out[base + lane] = dre * dre + dim * dim;
}

extern "C" void kernel_launch(void* const* d_in, const int* in_sizes, int n_in,
                              void* d_out, int out_size, void* d_ws, size_t ws_size,
                              hipStream_t stream)
{
  const float* x1 = (const float*)d_in[0];
  const float* x2 = (const float*)d_in[1];
  const float* W1 = (const float*)d_in[2];
  const float* b1 = (const float*)d_in[3];
  const float* W2 = (const float*)d_in[4];
  const float* b2 = (const float*)d_in[5];
  const float* W3 = (const float*)d_in[6];
  const float* b3 = (const float*)d_in[7];
  float* out = (float*)d_out;

  const int B = in_sizes[0] / 4;           // rows
  const int rowsPerBlock = 256;            // 8 wave32 * 32 rows
  dim3 block(256);
  dim3 grid((B + rowsPerBlock - 1) / rowsPerBlock);
  fidelity_kernel<<<grid, block, 0, stream>>>(x1, x2, W1, b1, W2, b2, W3, b3,
                                              out, B);
}